// DeterministicLattice_43508018709032
// MI455X (gfx1250) — compile-verified
//
#include <hip/hip_runtime.h>
#include <stdint.h>

// GenesisGeometry streaming stencil for MI455X (gfx1250).
// Memory-bound: ~775 MB moved per call -> ~33us floor at 23.3 TB/s.
// Uses CDNA5 async global->LDS (ASYNCcnt) input staging, v_tanh_f32
// transcendentals, and nontemporal (TH=NT) output stores.

typedef float v4f __attribute__((ext_vector_type(4)));
typedef int   v4i __attribute__((vector_size(16)));   // matches builtin param type

constexpr int   kN    = 65536;
constexpr int   kD    = 512;
constexpr int   kRows = 32;          // rows per block
constexpr int   kTPB  = kD / 4;      // 128 threads * float4 = one full row
constexpr float kPhi        = 1.61803398874989484820f;
constexpr float kInvPhiP2   = 1.0f / (kPhi + 2.0f);   // 1/(phi+2)
constexpr float kInvPhi     = 1.0f / kPhi;            // 1/phi

#if defined(__HIP_DEVICE_COMPILE__) && \
    __has_builtin(__builtin_amdgcn_global_load_async_to_lds_b128) && \
    __has_builtin(__builtin_amdgcn_s_wait_asynccnt)
#define USE_ASYNC_LDS 1
#else
#define USE_ASYNC_LDS 0
#endif

__device__ __forceinline__ float fast_tanh(float x) {
#if defined(__HIP_DEVICE_COMPILE__) && __has_builtin(__builtin_amdgcn_tanhf)
    return __builtin_amdgcn_tanhf(x);          // v_tanh_f32 (new on CDNA5)
#else
    float e = __expf(2.0f * x);                // v_exp_f32 fallback
    return (e - 1.0f) / (e + 1.0f);
#endif
}

#if USE_ASYNC_LDS
typedef __attribute__((address_space(1))) v4i* gv4i_ptr;
typedef __attribute__((address_space(3))) v4i* lv4i_ptr;

__device__ __forceinline__ void async_cp16(void* lds_dst, const void* gsrc) {
    // global_load_async_to_lds_b128: each lane copies 16B from its global
    // address to its LDS address; tracked by ASYNCcnt.
    gv4i_ptr g = (gv4i_ptr)(uintptr_t)gsrc;
    lv4i_ptr l = (lv4i_ptr)(uint32_t)(uintptr_t)lds_dst;
    __builtin_amdgcn_global_load_async_to_lds_b128(g, l, 0, 0);
}
#endif

__global__ __launch_bounds__(kTPB)
void genesis_geometry_kernel(const float* __restrict__ in, float* __restrict__ out) {
    const int    col   = threadIdx.x * 4;           // 16B-aligned column
    const int    r0    = blockIdx.x * kRows;
    const size_t plane = (size_t)kN * (size_t)kD;   // per-output stride

#if USE_ASYNC_LDS
    __shared__ float lbuf[2][kD];                   // double buffer, 4 KB
#endif

    // rolling window registers: xm = row i-1, xc = row i
    const int rm = (r0 == 0) ? (kN - 1) : (r0 - 1);
    v4f xm = *(const v4f*)(in + (size_t)rm * kD + col);
    v4f xc = *(const v4f*)(in + (size_t)r0 * kD + col);

#if USE_ASYNC_LDS
    {   // prime the pipeline: row r0+1 -> buffer 0
        int n0 = r0 + 1; if (n0 >= kN) n0 -= kN;
        async_cp16(&lbuf[0][col], in + (size_t)n0 * kD + col);
    }
#endif

    for (int r = 0; r < kRows; ++r) {
        const int row = r0 + r;

#if USE_ASYNC_LDS
        // issue prefetch of row+2 into the other buffer, then retire the
        // oldest async op (our current buffer) before reading it from LDS.
        int n2 = row + 2; if (n2 >= kN) n2 -= kN;
        async_cp16(&lbuf[(r + 1) & 1][col], in + (size_t)n2 * kD + col);
        __builtin_amdgcn_s_wait_asynccnt(1);
        v4f xp = *(const v4f*)&lbuf[r & 1][col];    // ds_load_b128
#else
        int n1 = row + 1; if (n1 >= kN) n1 -= kN;
        v4f xp = *(const v4f*)(in + (size_t)n1 * kD + col);
#endif

        // fused = (phi*x + left + right) / (phi + 2)
        v4f f = (xc * kPhi + xm + xp) * kInvPhiP2;

        v4f ident, bloom, crown, triad, spiral;
#pragma unroll
        for (int k = 0; k < 4; ++k) {
            const float fk = f[k];
            const float pf = kPhi * fk;
            const float t  = fast_tanh(fk);
            ident[k]  = fk + kInvPhi * t;                    // contractive update
            bloom[k]  = fast_tanh(pf);
            crown[k]  = 0.5f + 0.5f * fast_tanh(0.5f * pf);  // sigmoid(phi*f)
            triad[k]  = __sinf(fk) * __cosf(pf);             // v_sin / v_cos
            spiral[k] = fk * __expf(-__builtin_fabsf(fk) * kInvPhi); // v_exp
        }

        // streamed outputs: [identity, bloom, crown, triad, spiral]
        float* o = out + (size_t)row * kD + col;
        __builtin_nontemporal_store(ident,  (v4f*)(o));
        __builtin_nontemporal_store(bloom,  (v4f*)(o + plane));
        __builtin_nontemporal_store(crown,  (v4f*)(o + 2 * plane));
        __builtin_nontemporal_store(triad,  (v4f*)(o + 3 * plane));
        __builtin_nontemporal_store(spiral, (v4f*)(o + 4 * plane));

        xm = xc;
        xc = xp;
    }
}

extern "C" void kernel_launch(void* const* d_in, const int* in_sizes, int n_in,
                              void* d_out, int out_size, void* d_ws, size_t ws_size,
                              hipStream_t stream) {
    (void)in_sizes; (void)n_in; (void)out_size; (void)d_ws; (void)ws_size;
    const float* x = (const float*)d_in[0];
    float*     out = (float*)d_out;
    dim3 grid(kN / kRows);   // 2048 blocks
    dim3 block(kTPB);        // 128 threads (4 waves on wave32)
    genesis_geometry_kernel<<<grid, block, 0, stream>>>(x, out);
}